// Anchor_Loss2_76991583748729
// MI455X (gfx1250) — compile-verified
//
#include <hip/hip_runtime.h>

// Problem constants (from reference setup_inputs)
#define N_ROWS 65536
#define DIM    1024
#define NCLS   1000
#define NT     63            // 16x16 tiles per side: 63*16 = 1008 >= 1000
#define CP     (NT * 16)     // padded class count = 1008

// Workspace layout (in floats)
#define SX_OFF    0                      // CP * DIM   segment sums Sx[l][k]
#define CNT_OFF   (CP * DIM)             // CP         per-label counts
#define S2_OFF    (CNT_OFF + CP)         // CP         per-label sum ||x||^2
#define A2_OFF    (S2_OFF + CP)          // CP         per-anchor ||a||^2
#define SUMG_OFF  (A2_OFF + CP)          // CP         row sums of G = Sx @ A^T
#define GD_OFF    (SUMG_OFF + CP)        // CP         diagonal G[l][l]
#define SUMA2_OFF (GD_OFF + CP)          // 1          sum_j a2[j]
#define WS_FLOATS (SUMA2_OFF + 1)

typedef float v2f __attribute__((ext_vector_type(2)));
typedef float v8f __attribute__((ext_vector_type(8)));

__device__ __forceinline__ void atomAddF(float* p, float v) {
    // relaxed / agent scope -> no-return global_atomic_add_f32
    __hip_atomic_fetch_add(p, v, __ATOMIC_RELAXED, __HIP_MEMORY_SCOPE_AGENT);
}

// ---------------------------------------------------------------------------
// Kernel 1: segment sums over x.  One wave32 per row.
//   Sx[y_i] += x_i  (vectorized float4 loads, per-element f32 atomics into L2)
//   S2[y_i] += ||x_i||^2 ; cnt[y_i] += 1  (wave-reduced, lane 0 atomics)
// ---------------------------------------------------------------------------
__global__ void seg_sum_kernel(const float* __restrict__ x,
                               const int* __restrict__ y,
                               float* __restrict__ ws) {
    float* Sx  = ws + SX_OFF;
    float* cnt = ws + CNT_OFF;
    float* S2  = ws + S2_OFF;

    const int lane  = threadIdx.x & 31;
    const int wave  = blockIdx.x * (blockDim.x >> 5) + (threadIdx.x >> 5);
    const int nwave = gridDim.x * (blockDim.x >> 5);

    for (int row = wave; row < N_ROWS; row += nwave) {
        const int lbl = y[row];
        const float* xr = x + (size_t)row * DIM;
        float* dst = Sx + (size_t)lbl * DIM;
        float sq = 0.0f;
#pragma unroll
        for (int w = 0; w < DIM / 128; ++w) {
            const int col = (w * 32 + lane) * 4;        // coalesced 512B segments
            const float4 v = *(const float4*)(xr + col);
            sq += v.x * v.x + v.y * v.y + v.z * v.z + v.w * v.w;
            atomAddF(dst + col + 0, v.x);
            atomAddF(dst + col + 1, v.y);
            atomAddF(dst + col + 2, v.z);
            atomAddF(dst + col + 3, v.w);
        }
#pragma unroll
        for (int off = 16; off > 0; off >>= 1) sq += __shfl_xor(sq, off, 32);
        if (lane == 0) {
            atomAddF(S2 + lbl, sq);
            atomAddF(cnt + lbl, 1.0f);
        }
    }
}

// ---------------------------------------------------------------------------
// Kernel 2: a2[j] = ||anchors[j]||^2 and suma2 = sum_j a2[j].  Wave per row.
// ---------------------------------------------------------------------------
__global__ void a2_kernel(const float* __restrict__ anchors,
                          float* __restrict__ ws) {
    float* a2    = ws + A2_OFF;
    float* suma2 = ws + SUMA2_OFF;

    const int lane  = threadIdx.x & 31;
    const int wave  = blockIdx.x * (blockDim.x >> 5) + (threadIdx.x >> 5);
    const int nwave = gridDim.x * (blockDim.x >> 5);

    for (int row = wave; row < NCLS; row += nwave) {
        const float* ar = anchors + (size_t)row * DIM;
        float sq = 0.0f;
#pragma unroll
        for (int w = 0; w < DIM / 128; ++w) {
            const int col = (w * 32 + lane) * 4;
            const float4 v = *(const float4*)(ar + col);
            sq += v.x * v.x + v.y * v.y + v.z * v.z + v.w * v.w;
        }
#pragma unroll
        for (int off = 16; off > 0; off >>= 1) sq += __shfl_xor(sq, off, 32);
        if (lane == 0) {
            a2[row] = sq;
            atomAddF(suma2, sq);
        }
    }
}

// ---------------------------------------------------------------------------
// Kernel 3: G = Sx @ anchors^T via V_WMMA_F32_16X16X4_F32, one 16x16 tile per
// wave, K = 1024 in steps of 4.  Fused epilogue extracts G[l,l] and the row
// sums sum_j G[l,j] (j < NCLS) so G is never materialized.
//
// Fragment layout per ISA 7.12.2 (32-bit, wave32):
//   A (16x4, MxK): lane L holds row M=L&15, VGPRs {k0, k0+1}, k0=(L>=16)*2
//   B (4x16, KxN): lane L holds col N=L&15, VGPRs {k0, k0+1}
//   C/D (16x16):   VGPR r, lanes 0-15 -> (M=r, N=lane); lanes 16-31 -> (M=8+r, N=lane-16)
// ---------------------------------------------------------------------------
__global__ void gemm_wmma_kernel(const float* __restrict__ anchors,
                                 float* __restrict__ ws) {
    const float* Sx = ws + SX_OFF;
    float* sumG = ws + SUMG_OFF;
    float* Gd   = ws + GD_OFF;

    const int lane = threadIdx.x & 31;
    const int wave = blockIdx.x * (blockDim.x >> 5) + (threadIdx.x >> 5);
    if (wave >= NT * NT) return;                 // wave-uniform: EXEC stays all-1s for WMMA

    const int lt = wave / NT;                    // label-tile (rows of G)
    const int jt = wave % NT;                    // anchor-tile (cols of G)

    const int m  = lane & 15;
    const int k0 = (lane >> 4) * 2;
    const int row_l = lt * 16 + m;               // < CP; Sx padded rows are zero
    const int col_j = jt * 16 + m;
    const int jc    = (col_j < NCLS) ? col_j : (NCLS - 1);   // clamp OOB anchor reads

    const float* Ar = Sx      + (size_t)row_l * DIM + k0;
    const float* Br = anchors + (size_t)jc    * DIM + k0;

    v8f c = {};
#pragma unroll 8
    for (int k = 0; k < DIM; k += 4) {
        const v2f a = *(const v2f*)(Ar + k);
        const v2f b = *(const v2f*)(Br + k);
        c = __builtin_amdgcn_wmma_f32_16x16x4_f32(
                /*neg_a=*/false, a, /*neg_b=*/false, b,
                /*c_mod=*/(short)0, c, /*reuse_a=*/false, /*reuse_b=*/false);
    }

    const int base = lt * 16;

    // ---- Diagonal extraction (before column masking), branch-minimal. ----
    // Lanes 0..7  hold diag elements (M=N=lane)    in c[lane].
    // Lanes 24..31 hold diag elements (M=N=lane-16) in c[lane-24].
    if (lt == jt) {
        const int dr = (lane < 8) ? lane : (lane - 24);
        float dv = c[0];
#pragma unroll
        for (int r = 1; r < 8; ++r) dv = (dr == r) ? c[r] : dv;   // cndmask chain
        const bool holds = (lane < 8) | (lane >= 24);
        const int  l     = base + (lane & 15);
        if (holds && (l < NCLS)) Gd[l] = dv;
    }

    // ---- Row sums: mask padded columns, reduce each half's 16 lanes. ----
    const bool jvalid = (col_j < NCLS);
    float s[8];
#pragma unroll
    for (int r = 0; r < 8; ++r) {
        float v = jvalid ? c[r] : 0.0f;
        v += __shfl_xor(v, 1, 32);
        v += __shfl_xor(v, 2, 32);
        v += __shfl_xor(v, 4, 32);
        v += __shfl_xor(v, 8, 32);               // halves reduce independently
        s[r] = v;
    }
    if (lane == 0) {                             // rows M = 0..7
#pragma unroll
        for (int r = 0; r < 8; ++r) atomAddF(sumG + base + r, s[r]);
    } else if (lane == 16) {                     // rows M = 8..15
#pragma unroll
        for (int r = 0; r < 8; ++r) atomAddF(sumG + base + 8 + r, s[r]);
    }
}

// ---------------------------------------------------------------------------
// Kernel 4: finalize.
//   M[l,j] = (S2_l/c - 2 G[l,j]/c + a2_j)/D
//   loss   = sum_{l: cnt>0} ( 2 M[l,l] - sum_j M[l,j] )
// ---------------------------------------------------------------------------
__global__ void finalize_kernel(const float* __restrict__ ws,
                                float* __restrict__ out) {
    __shared__ float red[256];
    const float* cnt  = ws + CNT_OFF;
    const float* S2   = ws + S2_OFF;
    const float* a2   = ws + A2_OFF;
    const float* sumG = ws + SUMG_OFF;
    const float* Gd   = ws + GD_OFF;
    const float suma2 = ws[SUMA2_OFF];

    float acc = 0.0f;
    for (int l = (int)threadIdx.x; l < NCLS; l += (int)blockDim.x) {
        const float n = cnt[l];
        if (n > 0.0f) {
            const float inv = 1.0f / n;
            const float Mll = (S2[l] * inv - 2.0f * Gd[l] * inv + a2[l]) / (float)DIM;
            const float rowsum =
                ((float)NCLS * S2[l] * inv - 2.0f * sumG[l] * inv + suma2) / (float)DIM;
            acc += 2.0f * Mll - rowsum;
        }
    }
    red[threadIdx.x] = acc;
    __syncthreads();
    for (int s = 128; s > 0; s >>= 1) {
        if ((int)threadIdx.x < s) red[threadIdx.x] += red[threadIdx.x + s];
        __syncthreads();
    }
    if (threadIdx.x == 0) out[0] = red[0];
}

// ---------------------------------------------------------------------------
extern "C" void kernel_launch(void* const* d_in, const int* in_sizes, int n_in,
                              void* d_out, int out_size, void* d_ws, size_t ws_size,
                              hipStream_t stream) {
    (void)in_sizes; (void)n_in; (void)out_size; (void)ws_size;
    const float* x       = (const float*)d_in[0];
    const float* anchors = (const float*)d_in[1];
    const int*   y       = (const int*)d_in[2];   // JAX w/o x64 -> int32 labels
    float* out = (float*)d_out;
    float* ws  = (float*)d_ws;

    // Zero accumulators (capturable memset node).
    hipMemsetAsync(d_ws, 0, (size_t)WS_FLOATS * sizeof(float), stream);

    // 1) Segment sums over x: 4096 waves, 16 rows each.
    seg_sum_kernel<<<512, 256, 0, stream>>>(x, y, ws);

    // 2) Anchor norms.
    a2_kernel<<<32, 256, 0, stream>>>(anchors, ws);

    // 3) 1008x1008x1024 f32 WMMA GEMM, one tile per wave (3969 tiles, 8 waves/block).
    gemm_wmma_kernel<<<(NT * NT + 7) / 8, 256, 0, stream>>>(anchors, ws);

    // 4) Scalar loss.
    finalize_kernel<<<1, 256, 0, stream>>>(ws, out);
}